// APPNPEncoder_82162724372663
// MI455X (gfx1250) — compile-verified
//
#include <hip/hip_runtime.h>
#include <hip/hip_bf16.h>
#include <stdint.h>

typedef float v2f __attribute__((ext_vector_type(2)));
typedef float v8f __attribute__((ext_vector_type(8)));

#define N_NODES 100000
#define N_EDGES 1600000
#define IN_DIM  256
#define OUT_DIM 32
#define K_ITERS 10
#define ALPHA_F 0.2f
#define BETA_F  0.8f

// ---------------------------------------------------------------------------
// GEMM: h = x @ W^T + b  (also writes z0 = h, the initial PPR state)
// One wave per 16-row tile; computes BOTH 16-wide output column tiles
// (OUT_DIM = 32) with shared A operands. f32 WMMA 16x16x4, K = 256 -> 64
// steps x 2 tiles = 128 v_wmma_f32_16x16x4_f32 per wave.
// ---------------------------------------------------------------------------
__global__ __launch_bounds__(256) void appnp_gemm_wmma(
    const float* __restrict__ x, const float* __restrict__ W,
    const float* __restrict__ bias, float* __restrict__ h,
    float* __restrict__ z0) {
  __shared__ float Wl[OUT_DIM * IN_DIM];  // 32 KB of 320 KB LDS
  const int tid = threadIdx.x;

  // cooperative stage of W into LDS: 8192 floats, 256 threads, float4 x8 each
  {
    const float4* Wv = (const float4*)W;
    float4* Wlv = (float4*)Wl;
#pragma unroll
    for (int i = 0; i < 8; ++i) Wlv[tid + 256 * i] = Wv[tid + 256 * i];
  }
  __syncthreads();

  const int wave = tid >> 5;
  const int lane = tid & 31;
  const int tile = blockIdx.x * 8 + wave;  // 16-row tile index (wave-uniform)
  if (tile >= N_NODES / 16) return;        // uniform per wave -> EXEC stays all-1s

  const int m0 = tile * 16;
  const int c  = lane & 15;  // A: row-in-tile; B: column-in-tile; D: N index
  const int hh = lane >> 4;  // lane-half selects K pair (ISA 7.12.2, 32-bit)

  const float* xrow = x + (size_t)(m0 + c) * IN_DIM;  // A row (two lanes/row)
  const float* w0 = Wl + c * IN_DIM;                  // B column n = c
  const float* w1 = Wl + (16 + c) * IN_DIM;           // B column n = 16 + c

  v8f acc0 = {0.f, 0.f, 0.f, 0.f, 0.f, 0.f, 0.f, 0.f};
  v8f acc1 = {0.f, 0.f, 0.f, 0.f, 0.f, 0.f, 0.f, 0.f};

#pragma unroll 8
  for (int kb = 0; kb < IN_DIM; kb += 4) {
    const int kk = kb + 2 * hh;  // lanes 0-15: K={kb,kb+1}; 16-31: {kb+2,kb+3}
    v2f a  = *(const v2f*)(xrow + kk);
    v2f b0 = *(const v2f*)(w0 + kk);
    v2f b1 = *(const v2f*)(w1 + kk);
    // (neg_a, A, neg_b, B, c_mod, C, reuse_a, reuse_b)
    acc0 = __builtin_amdgcn_wmma_f32_16x16x4_f32(false, a, false, b0,
                                                 (short)0, acc0, false, false);
    acc1 = __builtin_amdgcn_wmma_f32_16x16x4_f32(false, a, false, b1,
                                                 (short)0, acc1, false, false);
  }

  const float b0v = bias[c];
  const float b1v = bias[16 + c];
#pragma unroll
  for (int r = 0; r < 8; ++r) {
    // D layout: VGPR r, lanes 0-15 -> M = r; lanes 16-31 -> M = r + 8; N = c
    const int row = m0 + r + 8 * hh;
    const float v0 = acc0[r] + b0v;
    const float v1 = acc1[r] + b1v;
    h[row * OUT_DIM + c]       = v0;
    h[row * OUT_DIM + 16 + c]  = v1;
    z0[row * OUT_DIM + c]      = v0;
    z0[row * OUT_DIM + 16 + c] = v1;
  }
}

// ---------------------------------------------------------------------------
// Degree / normalization kernels
// ---------------------------------------------------------------------------
__global__ void appnp_deg_init(float* __restrict__ deg) {
  const int n = blockIdx.x * blockDim.x + threadIdx.x;
  if (n < N_NODES) deg[n] = 1.0f;  // self-loop contribution
}

__global__ void appnp_deg_count(const long long* __restrict__ ei,
                                float* __restrict__ deg) {
  const int e = blockIdx.x * blockDim.x + threadIdx.x;
  if (e < N_EDGES) atomicAdd(&deg[(int)ei[N_EDGES + e]], 1.0f);  // col = target
}

__global__ void appnp_dinv(float* __restrict__ d) {
  const int n = blockIdx.x * blockDim.x + threadIdx.x;
  if (n < N_NODES) {
    const float v = d[n];
    d[n] = (v > 0.f) ? rsqrtf(v) : 0.f;  // in-place deg -> deg^{-1/2}
  }
}

__global__ void appnp_norm(const long long* __restrict__ ei,
                           const float* __restrict__ dinv,
                           float* __restrict__ norm) {
  const int e = blockIdx.x * blockDim.x + threadIdx.x;
  if (e < N_EDGES)
    norm[e] = dinv[(int)ei[e]] * dinv[(int)ei[N_EDGES + e]];
}

// ---------------------------------------------------------------------------
// One PPR step: z_new = beta * (A_hat z_old) + alpha * h
//   init kernel folds in the self-loop term: beta * dinv[n]^2 * z_old + alpha*h
//   edge kernel scatter-adds beta * norm[e] * z_old[row] into z_new[col]
//   (z fits in L2: 12.8 MB; this phase is L2-atomic-bandwidth bound)
// ---------------------------------------------------------------------------
__global__ void appnp_prop_init(const float* __restrict__ h,
                                const float* __restrict__ zold,
                                const float* __restrict__ dinv,
                                float* __restrict__ znew) {
  const int idx = blockIdx.x * blockDim.x + threadIdx.x;
  if (idx < N_NODES * OUT_DIM) {
    const float di = dinv[idx >> 5];
    znew[idx] = ALPHA_F * h[idx] + BETA_F * di * di * zold[idx];
  }
}

__global__ void appnp_prop_edge(const long long* __restrict__ ei,
                                const float* __restrict__ norm,
                                const float* __restrict__ zold,
                                float* __restrict__ znew) {
  const int idx = blockIdx.x * blockDim.x + threadIdx.x;  // < 51.2M
  const int e = idx >> 5;   // one wave (32 lanes) per edge
  const int c = idx & 31;   // one lane per channel -> 128B coalesced
  if (e < N_EDGES) {
    const int row = (int)ei[e];
    const int col = (int)ei[N_EDGES + e];
    const float w = BETA_F * norm[e];
    atomicAdd(&znew[col * OUT_DIM + c], w * zold[row * OUT_DIM + c]);
  }
}

__global__ void appnp_prelu(const float* __restrict__ z,
                            const float* __restrict__ a,
                            float* __restrict__ out) {
  const int idx = blockIdx.x * blockDim.x + threadIdx.x;
  if (idx < N_NODES * OUT_DIM) {
    const float v = z[idx];
    out[idx] = (v >= 0.f) ? v : a[idx & 31] * v;
  }
}

// ---------------------------------------------------------------------------
extern "C" void kernel_launch(void* const* d_in, const int* in_sizes, int n_in,
                              void* d_out, int out_size, void* d_ws,
                              size_t ws_size, hipStream_t stream) {
  const float* x        = (const float*)d_in[0];
  const long long* ei   = (const long long*)d_in[1];  // int64 [2, E]
  const float* W        = (const float*)d_in[2];
  const float* bias     = (const float*)d_in[3];
  const float* prelu_a  = (const float*)d_in[4];
  float* out            = (float*)d_out;

  // workspace carve-out (~45 MB)
  float* h    = (float*)d_ws;                 // N*32
  float* z0   = h    + (size_t)N_NODES * OUT_DIM;
  float* z1   = z0   + (size_t)N_NODES * OUT_DIM;
  float* dinv = z1   + (size_t)N_NODES * OUT_DIM;  // N (deg, then deg^-1/2)
  float* norm = dinv + N_NODES;                    // E

  const int TB = 256;
  const int gN  = (N_NODES + TB - 1) / TB;
  const int gE  = (N_EDGES + TB - 1) / TB;
  const int gNC = (N_NODES * OUT_DIM + TB - 1) / TB;
  const int gEC = (N_EDGES * 32 + TB - 1) / TB;

  // degree + normalization
  appnp_deg_init<<<gN, TB, 0, stream>>>(dinv);
  appnp_deg_count<<<gE, TB, 0, stream>>>(ei, dinv);
  appnp_dinv<<<gN, TB, 0, stream>>>(dinv);
  appnp_norm<<<gE, TB, 0, stream>>>(ei, dinv, norm);

  // projection via f32 WMMA (6250 row-tiles, 8 waves/block)
  appnp_gemm_wmma<<<(6250 + 7) / 8, TB, 0, stream>>>(x, W, bias, h, z0);

  // K power-iteration steps, ping-pong z buffers
  float* zc = z0;
  float* zn = z1;
  for (int it = 0; it < K_ITERS; ++it) {
    appnp_prop_init<<<gNC, TB, 0, stream>>>(h, zc, dinv, zn);
    appnp_prop_edge<<<gEC, TB, 0, stream>>>(ei, norm, zc, zn);
    float* t = zc; zc = zn; zn = t;
  }

  appnp_prelu<<<gNC, TB, 0, stream>>>(zc, prelu_a, out);
}